// EvolvedNet_80032420593868
// MI455X (gfx1250) — compile-verified
//
#include <hip/hip_runtime.h>
#include <math.h>

#define N_NODES   32
#define N_INPUTS  8
#define N_OUTPUTS 4
#define N_EDGES   128
#define BATCH     524288

#define T    128            // threads per block
#define ILP  4              // batch elements per thread
#define SLOTS (T * ILP)     // 512 batch elements per block; LDS row = 2048 B

// ---- CDNA5 hardware tanh (v_tanh_f32) with libm fallback ------------------
__device__ __forceinline__ float hw_tanh(float x) {
#if defined(__has_builtin)
#if __has_builtin(__builtin_amdgcn_tanhf)
  return __builtin_amdgcn_tanhf(x);
#elif __has_builtin(__builtin_amdgcn_tanh_f32)
  return __builtin_amdgcn_tanh_f32(x);
#else
  return tanhf(x);
#endif
#else
  return tanhf(x);
#endif
}

// ---- optional async global->LDS staging (gfx1250) -------------------------
#if defined(__has_builtin)
#if __has_builtin(__builtin_amdgcn_global_load_async_to_lds_b128) && \
    __has_builtin(__builtin_amdgcn_s_wait_asynccnt)
#define USE_ASYNC_LDS 1
#endif
#endif
#ifndef USE_ASYNC_LDS
#define USE_ASYNC_LDS 0
#endif

typedef float f32x4 __attribute__((vector_size(16)));
typedef int   v4i   __attribute__((vector_size(16)));
typedef __attribute__((address_space(1))) v4i* gbl_v4i_p;  // global AS pointer
typedef __attribute__((address_space(3))) v4i* lds_v4i_p;  // LDS AS pointer

__global__ __launch_bounds__(T) void evonet_kernel(
    const float* __restrict__ x,    // [N_INPUTS, BATCH]
    const float* __restrict__ w,    // [N_EDGES]
    const int*   __restrict__ src,  // [N_EDGES]
    const int*   __restrict__ dst,  // [N_EDGES]
    float*       __restrict__ out)  // [N_OUTPUTS, BATCH]
{
  // vals[node][slot]; slot = i*T + tid  (lane-consecutive -> bank-conflict free)
  __shared__ float vals[N_NODES * SLOTS];           // 64 KB

  const int tid       = threadIdx.x;
  const int blockBase = blockIdx.x * SLOTS;         // first batch element of block

  // ---------------- init: rows 0..7 <- x, rows 8..31 <- 0 ----------------
  // Each LDS row (2048 B) is a contiguous copy of x[n, blockBase : blockBase+512].
#if USE_ASYNC_LDS
  {
    // 128 lanes x 16 B = 2048 B per row; lane tid copies bytes [16*tid, 16*tid+16)
    #pragma unroll
    for (int n = 0; n < N_INPUTS; ++n) {
      float* g = const_cast<float*>(x) + (size_t)n * BATCH + blockBase + 4 * tid;
      float* l = &vals[n * SLOTS + 4 * tid];
      __builtin_amdgcn_global_load_async_to_lds_b128(
          (gbl_v4i_p)g,            // addrspacecast generic -> global(1)
          (lds_v4i_p)l,            // addrspacecast generic -> lds(3)
          /*offset=*/0, /*cpol=*/0);
    }
  }
#else
  {
    #pragma unroll
    for (int n = 0; n < N_INPUTS; ++n) {
      const f32x4 v = *(const f32x4*)(x + (size_t)n * BATCH + blockBase + 4 * tid);
      *(f32x4*)&vals[n * SLOTS + 4 * tid] = v;
    }
  }
#endif
  {
    const f32x4 z = {0.f, 0.f, 0.f, 0.f};
    #pragma unroll
    for (int n = N_INPUTS; n < N_NODES; ++n)
      *(f32x4*)&vals[n * SLOTS + 4 * tid] = z;
  }
#if USE_ASYNC_LDS
  __builtin_amdgcn_s_wait_asynccnt(0);
#endif
  __syncthreads();  // async-copied rows are consumed by other threads' slots

  // ---------------- relaxation sweeps (faithful sequential edges) ----------
  float* my = &vals[tid];   // value of (node n, copy i) at my[n*SLOTS + i*T]

  for (int sweep = 0; sweep < N_NODES; ++sweep) {
    #pragma unroll 4
    for (int e = 0; e < N_EDGES; ++e) {
      const int   s  = src[e];   // uniform -> s_load
      const int   d  = dst[e];   // uniform -> s_load
      const float wv = w[e];     // uniform -> s_load

      const float* sp = &my[s * SLOTS];   // base VGPR + uniform SGPR offset
      float*       dp = &my[d * SLOTS];

      #pragma unroll
      for (int i = 0; i < ILP; ++i) {
        const float t = hw_tanh(sp[i * T] * wv);        // ds_load + v_mul + v_tanh
        // ds_add_f32 (no return); per-wave DS ops are in-order, preserving the
        // sequential edge semantics for each thread's private slots.
        __hip_atomic_fetch_add(&dp[i * T], t,
                               __ATOMIC_RELAXED, __HIP_MEMORY_SCOPE_WORKGROUP);
      }
    }
  }

  // ---------------- outputs: tanh(vals[28..31]) ---------------------------
  #pragma unroll
  for (int o = 0; o < N_OUTPUTS; ++o) {
    const float* op = &my[(N_NODES - N_OUTPUTS + o) * SLOTS];
    #pragma unroll
    for (int i = 0; i < ILP; ++i) {
      out[(size_t)o * BATCH + blockBase + i * T + tid] = hw_tanh(op[i * T]);
    }
  }
}

extern "C" void kernel_launch(void* const* d_in, const int* in_sizes, int n_in,
                              void* d_out, int out_size, void* d_ws, size_t ws_size,
                              hipStream_t stream) {
  (void)in_sizes; (void)n_in; (void)out_size; (void)d_ws; (void)ws_size;
  const float* x   = (const float*)d_in[0];
  const float* w   = (const float*)d_in[1];
  const int*   src = (const int*)d_in[2];
  const int*   dst = (const int*)d_in[3];
  float*       out = (float*)d_out;

  const int blocks = BATCH / SLOTS;   // 524288 / 512 = 1024
  evonet_kernel<<<blocks, T, 0, stream>>>(x, w, src, dst, out);
}